// DependencyParser_44298292691161
// MI455X (gfx1250) — compile-verified
//
#include <hip/hip_runtime.h>
#include <hip/hip_bf16.h>
#include <math.h>

// Problem constants (from reference setup_inputs)
#define BB 4
#define SS 256
#define HH 768
#define LL 4
#define KK 4              // top-k
#define H2 (2 * HH)       // 1536
#define M1 (BB * SS)      // 1024 rows for h_i/h_j GEMM
#define M2 (BB * SS * KK) // 4096 rows for dense GEMM

#define BPAD 68           // LDS row stride (dwords) for 64-dword tiles + 4 pad

typedef __attribute__((ext_vector_type(2))) float v2f;
typedef __attribute__((ext_vector_type(8))) float v8f;
typedef __attribute__((ext_vector_type(4))) unsigned int v4u;
typedef __attribute__((ext_vector_type(8))) int v8i;
typedef __attribute__((ext_vector_type(4))) int v4i;

// ---------------------------------------------------------------------------
// Tensor Data Mover: 2D tile load Global -> LDS (D# per CDNA5 ISA ch.8).
// Group0: count=1 | lds_addr | global_addr[56:0] | type=2.
// Group1: data_size=4B, optional LDS padding (pad_interval code 5 = 64 dwords,
//         pad_amount code 3 = 4 dwords -> row stride 68 dwords), tile/tensor
//         dims, tensor_dim0_stride. Groups 2/3 zero (<=2D tensor).
// This toolchain exposes the 6-arg builtin:
//   (uint32x4 g0, int32x8 g1, int32x4, int32x4, int32x8, i32 cpol)
// ---------------------------------------------------------------------------
__device__ __forceinline__ void tdm_load_2d(unsigned int lds_byte_addr,
                                            const void* gptr,
                                            unsigned int tile_d0,    // f32 elems along dim0
                                            unsigned int tile_d1,    // rows
                                            unsigned int tensor_d0,  // bounds dim0 (elems)
                                            unsigned int tensor_d1,  // bounds dim1 (rows)
                                            unsigned int stride_d0,  // elems between rows
                                            unsigned int pad_enable,
                                            unsigned int pad_interval,
                                            unsigned int pad_amount)
{
    const unsigned long long ga = (unsigned long long)(uintptr_t)gptr;
    v4u g0;
    g0[0] = 1u;                                                    // count=1 (valid user D#)
    g0[1] = lds_byte_addr;                                         // lds_addr
    g0[2] = (unsigned int)(ga & 0xFFFFFFFFu);                      // global_addr[31:0]
    g0[3] = (unsigned int)((ga >> 32) & 0x1FFFFFFu) | (2u << 30);  // global_addr[56:32] | type=2
    v8i g1;
    g1[0] = (int)((2u << 16) | (pad_enable << 20) |
                  (pad_interval << 22) | (pad_amount << 25));      // data_size=4B + padding
    g1[1] = (int)((tensor_d0 & 0xFFFFu) << 16);                    // tensor_dim0[15:0]
    g1[2] = (int)(((tensor_d0 >> 16) & 0xFFFFu) |
                  ((tensor_d1 & 0xFFFFu) << 16));                  // tdim0[31:16] | tdim1[15:0]
    g1[3] = (int)(((tensor_d1 >> 16) & 0xFFFFu) |
                  ((tile_d0 & 0xFFFFu) << 16));                    // tdim1[31:16] | tile_dim0
    g1[4] = (int)(tile_d1 & 0xFFFFu);                              // tile_dim1 (tile_dim2=0)
    g1[5] = (int)stride_d0;                                        // tensor_dim0_stride[31:0]
    g1[6] = 0;                                                     // stride hi | dim1_stride lo
    g1[7] = 0;
    const v4i gz4 = {0, 0, 0, 0};
    const v8i gz8 = {0, 0, 0, 0, 0, 0, 0, 0};
    __builtin_amdgcn_tensor_load_to_lds(g0, g1, gz4, gz4, gz8, 0);
}

// ---------------------------------------------------------------------------
// Kernel 1: h_i = pooled @ Wa^T + Wa_b ; h_j = pooled @ Ua^T + Ua_b
// 8 waves/block share a TDM-staged 64x64 B tile in LDS; each wave computes a
// 16(M) x 64(N) tile via V_WMMA_F32_16X16X4_F32.
// A layout (16x4 f32): lane l, vgpr v -> (M=l&15, K=v+2*(l>>4)) = one float2.
// B layout (4x16):     lane l, vgpr v -> (K=v+2*(l>>4), N=l&15) = one ds b64.
// ---------------------------------------------------------------------------
__global__ void __launch_bounds__(256)
k_gemm_hij(const float* __restrict__ pooled,
           const float* __restrict__ Wa, const float* __restrict__ Wab,
           const float* __restrict__ Ua, const float* __restrict__ Uab,
           float* __restrict__ hi, float* __restrict__ hj)
{
    __shared__ float sB[64 * BPAD];
    const int lane = threadIdx.x & 31;
    const int wave = threadIdx.x >> 5;
    const int lmod = lane & 15;
    const int lgrp = lane >> 4;

    const float* W    = blockIdx.z ? Ua  : Wa;
    const float* bias = blockIdx.z ? Uab : Wab;
    float*       out  = blockIdx.z ? hj  : hi;

    const int m0 = (blockIdx.x * 8 + wave) * 16;   // 8 waves/block along M
    const int n0 = blockIdx.y * 64;                // 64 cols per block

    const float* aRow = pooled + (size_t)(m0 + lmod) * HH;

    v8f c0 = {}, c1 = {}, c2 = {}, c3 = {};
    for (int k0 = 0; k0 < HH; k0 += 64) {
        if (threadIdx.x == 0) {
            tdm_load_2d((unsigned int)(uintptr_t)sB,
                        W + (size_t)n0 * HH + k0,
                        64, 64, HH, HH, HH,
                        1u, 5u, 3u);               // pad: every 64 dwords add 4
            __builtin_amdgcn_s_wait_tensorcnt(0);
        }
        __syncthreads();
        const float* bBase = sB + 2 * lgrp;
#pragma unroll 4
        for (int kl = 0; kl < 64; kl += 4) {
            v2f a  = *(const v2f*)(aRow + k0 + kl + 2 * lgrp);
            v2f b0 = *(const v2f*)(bBase + (lmod + 0)  * BPAD + kl);
            v2f b1 = *(const v2f*)(bBase + (lmod + 16) * BPAD + kl);
            v2f b2 = *(const v2f*)(bBase + (lmod + 32) * BPAD + kl);
            v2f b3 = *(const v2f*)(bBase + (lmod + 48) * BPAD + kl);
            c0 = __builtin_amdgcn_wmma_f32_16x16x4_f32(false, a, false, b0, (short)0, c0, false, false);
            c1 = __builtin_amdgcn_wmma_f32_16x16x4_f32(false, a, false, b1, (short)0, c1, false, false);
            c2 = __builtin_amdgcn_wmma_f32_16x16x4_f32(false, a, false, b2, (short)0, c2, false, false);
            c3 = __builtin_amdgcn_wmma_f32_16x16x4_f32(false, a, false, b3, (short)0, c3, false, false);
        }
        __syncthreads();
    }
    // C/D layout: lane l, vgpr v -> (M = v + 8*(l>>4), N = l&15)
#pragma unroll
    for (int v = 0; v < 8; ++v) {
        const int mr = m0 + v + 8 * lgrp;
        float* o = out + (size_t)mr * HH + n0 + lmod;
        o[0]  = c0[v] + bias[n0 + lmod];
        o[16] = c1[v] + bias[n0 + 16 + lmod];
        o[32] = c2[v] + bias[n0 + 32 + lmod];
        o[48] = c3[v] + bias[n0 + 48 + lmod];
    }
}

// ---------------------------------------------------------------------------
// Kernel 2: dependency_logits[b,a,c] = sum_h va[h]*tanh(h_i[b,c,h]+h_j[b,a,h])
// Block per (a,b): h_j row + va TDM-staged in LDS; each wave owns 32 c-values,
// lanes stride over h (coalesced), wave32 shuffle reduction.
// ---------------------------------------------------------------------------
__global__ void __launch_bounds__(256)
k_scores(const float* __restrict__ hi, const float* __restrict__ hj,
         const float* __restrict__ va, float* __restrict__ logits)
{
    __shared__ float shj[HH];
    __shared__ float sva[HH];
    const int a = blockIdx.x;
    const int b = blockIdx.y;

    const float* hjrow = hj + ((size_t)b * SS + a) * HH;
    if (threadIdx.x == 0) {
        tdm_load_2d((unsigned int)(uintptr_t)shj, hjrow, HH, 1, HH, 1, HH, 0u, 0u, 0u);
        tdm_load_2d((unsigned int)(uintptr_t)sva, va,    HH, 1, HH, 1, HH, 0u, 0u, 0u);
        __builtin_amdgcn_s_wait_tensorcnt(0);
    }
    __syncthreads();

    const int lane = threadIdx.x & 31;
    const int wave = threadIdx.x >> 5;
    float* orow = logits + ((size_t)b * SS + a) * SS;

    for (int i = 0; i < 32; ++i) {
        const int c = wave * 32 + i;
        const float* hirow = hi + ((size_t)b * SS + c) * HH;
        float acc = 0.f;
#pragma unroll 4
        for (int h = lane; h < HH; h += 32)
            acc += sva[h] * tanhf(hirow[h] + shj[h]);
#pragma unroll
        for (int off = 16; off > 0; off >>= 1)
            acc += __shfl_down(acc, off, 32);
        if (lane == 0) orow[c] = acc;
    }
}

// ---------------------------------------------------------------------------
// Kernel 3: top-K (K=4) indices over axis c (256 values) per (b,s) row.
// Lowest-index-wins tie break matches jax.lax.top_k.
// ---------------------------------------------------------------------------
__global__ void __launch_bounds__(256)
k_topk(const float* __restrict__ logits, int* __restrict__ idx)
{
    const int m = blockIdx.x * blockDim.x + threadIdx.x; // (b*S + s)
    if (m >= M1) return;
    const float* row = logits + (size_t)m * SS;
    int chosen[KK];
#pragma unroll
    for (int k = 0; k < KK; ++k) {
        float best = -INFINITY;
        int bi = 0;
        for (int c = 0; c < SS; ++c) {
            bool skip = false;
#pragma unroll
            for (int j = 0; j < KK; ++j)
                if (j < k && chosen[j] == c) skip = true;
            const float v = row[c];
            if (!skip && v > best) { best = v; bi = c; }
        }
        chosen[k] = bi;
    }
#pragma unroll
    for (int k = 0; k < KK; ++k) idx[m * KK + k] = chosen[k];
}

// ---------------------------------------------------------------------------
// Kernel 4: hidden = tanh(cat @ dense_w^T + dense_b), cat = [dep || head]
// gathered on the fly. M2=4096, Kdim=1536, N=768. TDM-staged 64x64 weight
// chunks, WMMA f32 16x16x4, 16x64 output per wave. The float2 A-load never
// straddles the H boundary (columns 2-aligned, H even).
// ---------------------------------------------------------------------------
__global__ void __launch_bounds__(256)
k_dense(const float* __restrict__ pooled, const int* __restrict__ gidx,
        const float* __restrict__ dw, const float* __restrict__ db,
        float* __restrict__ hidden)
{
    __shared__ float sB[64 * BPAD];
    const int lane = threadIdx.x & 31;
    const int wave = threadIdx.x >> 5;
    const int lmod = lane & 15;
    const int lgrp = lane >> 4;

    const int m0 = (blockIdx.x * 8 + wave) * 16;
    const int n0 = blockIdx.y * 64;

    const int m  = m0 + lmod;         // row in (b,s,k) flattened order
    const int bs = m >> 2;            // b*S + s
    const int bq = bs >> 8;           // b  (S == 256)
    const int hd = gidx[m];           // head token index in [0,S)

    const float* depRow  = pooled + (size_t)bs * HH;
    const float* headRow = pooled + ((size_t)bq * SS + hd) * HH;

    v8f c0 = {}, c1 = {}, c2 = {}, c3 = {};
    for (int k0 = 0; k0 < H2; k0 += 64) {
        if (threadIdx.x == 0) {
            tdm_load_2d((unsigned int)(uintptr_t)sB,
                        dw + (size_t)n0 * H2 + k0,
                        64, 64, H2, HH, H2,
                        1u, 5u, 3u);
            __builtin_amdgcn_s_wait_tensorcnt(0);
        }
        __syncthreads();
        const float* bBase = sB + 2 * lgrp;
#pragma unroll 4
        for (int kl = 0; kl < 64; kl += 4) {
            const int col = k0 + kl + 2 * lgrp;
            const float* ap = (col < HH) ? (depRow + col) : (headRow + (col - HH));
            v2f a  = *(const v2f*)ap;
            v2f b0 = *(const v2f*)(bBase + (lmod + 0)  * BPAD + kl);
            v2f b1 = *(const v2f*)(bBase + (lmod + 16) * BPAD + kl);
            v2f b2 = *(const v2f*)(bBase + (lmod + 32) * BPAD + kl);
            v2f b3 = *(const v2f*)(bBase + (lmod + 48) * BPAD + kl);
            c0 = __builtin_amdgcn_wmma_f32_16x16x4_f32(false, a, false, b0, (short)0, c0, false, false);
            c1 = __builtin_amdgcn_wmma_f32_16x16x4_f32(false, a, false, b1, (short)0, c1, false, false);
            c2 = __builtin_amdgcn_wmma_f32_16x16x4_f32(false, a, false, b2, (short)0, c2, false, false);
            c3 = __builtin_amdgcn_wmma_f32_16x16x4_f32(false, a, false, b3, (short)0, c3, false, false);
        }
        __syncthreads();
    }
#pragma unroll
    for (int v = 0; v < 8; ++v) {
        const int mr = m0 + v + 8 * lgrp;
        float* o = hidden + (size_t)mr * HH + n0 + lmod;
        o[0]  = tanhf(c0[v] + db[n0 + lmod]);
        o[16] = tanhf(c1[v] + db[n0 + 16 + lmod]);
        o[32] = tanhf(c2[v] + db[n0 + 32 + lmod]);
        o[48] = tanhf(c3[v] + db[n0 + 48 + lmod]);
    }
}

// ---------------------------------------------------------------------------
// Kernel 5: dependency_type_logits = hidden @ fc_w^T + fc_b (N=4: VALU dots).
// One wave per row, lanes stride over H, shuffle-reduce 4 accumulators.
// ---------------------------------------------------------------------------
__global__ void __launch_bounds__(256)
k_fc(const float* __restrict__ hidden, const float* __restrict__ fw,
     const float* __restrict__ fb, float* __restrict__ out2)
{
    const int lane = threadIdx.x & 31;
    const int wave = threadIdx.x >> 5;
    const int m = blockIdx.x * 8 + wave;
    const float* hrow = hidden + (size_t)m * HH;

    float a0 = 0.f, a1 = 0.f, a2 = 0.f, a3 = 0.f;
#pragma unroll 4
    for (int h = lane; h < HH; h += 32) {
        const float v = hrow[h];
        a0 += v * fw[0 * HH + h];
        a1 += v * fw[1 * HH + h];
        a2 += v * fw[2 * HH + h];
        a3 += v * fw[3 * HH + h];
    }
#pragma unroll
    for (int off = 16; off > 0; off >>= 1) {
        a0 += __shfl_down(a0, off, 32);
        a1 += __shfl_down(a1, off, 32);
        a2 += __shfl_down(a2, off, 32);
        a3 += __shfl_down(a3, off, 32);
    }
    if (lane == 0) {
        out2[(size_t)m * LL + 0] = a0 + fb[0];
        out2[(size_t)m * LL + 1] = a1 + fb[1];
        out2[(size_t)m * LL + 2] = a2 + fb[2];
        out2[(size_t)m * LL + 3] = a3 + fb[3];
    }
}

// ---------------------------------------------------------------------------
extern "C" void kernel_launch(void* const* d_in, const int* in_sizes, int n_in,
                              void* d_out, int out_size, void* d_ws, size_t ws_size,
                              hipStream_t stream)
{
    const float* pooled = (const float*)d_in[0];
    const float* Wa_w   = (const float*)d_in[1];
    const float* Wa_b   = (const float*)d_in[2];
    const float* Ua_w   = (const float*)d_in[3];
    const float* Ua_b   = (const float*)d_in[4];
    const float* va_w   = (const float*)d_in[5];
    const float* dw     = (const float*)d_in[6];
    const float* db     = (const float*)d_in[7];
    const float* fw     = (const float*)d_in[8];
    const float* fb     = (const float*)d_in[9];

    float* logits = (float*)d_out;                        // (B,S,S)
    float* out2   = (float*)d_out + (size_t)BB * SS * SS; // (B,S,K,L)

    // Workspace layout: hi | hj | hidden | idx
    char* ws = (char*)d_ws;
    float* hi     = (float*)ws;                          // 1024*768
    float* hj     = hi + (size_t)M1 * HH;                // 1024*768
    float* hidden = hj + (size_t)M1 * HH;                // 4096*768
    int*   gidx   = (int*)(hidden + (size_t)M2 * HH);    // 4096

    // 1) h_i / h_j GEMMs (WMMA f32, TDM weight staging)
    dim3 g1(M1 / 128, HH / 64, 2);
    k_gemm_hij<<<g1, 256, 0, stream>>>(pooled, Wa_w, Wa_b, Ua_w, Ua_b, hi, hj);

    // 2) biaffine scores
    dim3 g2(SS, BB);
    k_scores<<<g2, 256, 0, stream>>>(hi, hj, va_w, logits);

    // 3) top-k head candidates
    k_topk<<<dim3(M1 / 256), 256, 0, stream>>>(logits, gidx);

    // 4) dense classifier with fused gather+concat (WMMA f32, TDM staging)
    dim3 g4(M2 / 128, HH / 64);
    k_dense<<<g4, 256, 0, stream>>>(pooled, gidx, dw, db, hidden);

    // 5) fc head
    k_fc<<<dim3(M2 / 8), 256, 0, stream>>>(hidden, fw, fb, out2);
}